// EDMLoss_59468117180629
// MI455X (gfx1250) — compile-verified
//
#include <hip/hip_runtime.h>
#include <math.h>

// Problem constants (from reference setup_inputs)
#define BB   8
#define TT   1024
#define CC   32
#define FF   256
#define DD   128
#define KK   512
#define BT   8192          // B*T

typedef __attribute__((ext_vector_type(2))) float v2f;
typedef __attribute__((ext_vector_type(8))) float v8f;

// Workspace layout (floats):
//  [0] sum (h - z)^2 over all (b,t,d)
//  [1] sum residual^2 over (b,t,c)
//  [2] sum Xhat . w_d over (b,t)
//  [3] pad
//  [4 .. 4+FF)        S[f]   = sum_rows Hdec[row][f]
//  [260 .. 260+CC*FF) E[c,f] = sum_rows R[row][c]*Hdec[row][f]
#define WS_S     4
#define WS_E     260
#define WS_TOTAL (WS_E + CC*FF)

// ---- CDNA5 async global->LDS copies (ASYNCcnt-tracked, no VGPR data path) --
__device__ __forceinline__ uint32_t lds_off(const void* p) {
  // flat-LDS aperture keeps the wave-relative LDS byte offset in addr[31:0]
  return (uint32_t)(uintptr_t)p;
}
__device__ __forceinline__ void async_b128(uint32_t ldsoff, const void* g) {
  asm volatile("global_load_async_to_lds_b128 %0, %1, off"
               :: "v"(ldsoff), "v"(g) : "memory");
}
__device__ __forceinline__ void async_b32(uint32_t ldsoff, const void* g) {
  asm volatile("global_load_async_to_lds_b32 %0, %1, off"
               :: "v"(ldsoff), "v"(g) : "memory");
}
__device__ __forceinline__ void wait_async0() {
  asm volatile("s_wait_asynccnt 0x0" ::: "memory");
}

__global__ void k_init(float* __restrict__ ws) {
  int i = blockIdx.x * 256 + threadIdx.x;
  if (i < WS_TOTAL) ws[i] = 0.0f;
}

// ---------------------------------------------------------------------------
// L1 nearest-codeword search + sum of squared L2 distance to the winner.
// One thread per (b,t). h column async-gathered into LDS; codebook streamed
// through a double-buffered LDS chunk (32 codewords x 128 dims, d-major so
// async B128 segments are contiguous on both sides). Prefetch of chunk kb+1
// overlaps the distance computation against chunk kb. Strict '<' over
// ascending k reproduces the reference tie-breaking exactly.
// ---------------------------------------------------------------------------
__global__ void k_mem(const float* __restrict__ H, const float* __restrict__ M,
                      float* __restrict__ ws) {
  __shared__ __attribute__((aligned(16))) float h_lds[64 * 132]; // 64 pos x 128 d (pad)
  __shared__ __attribute__((aligned(16))) float mbuf[2][DD * 32]; // [d][k] chunks
  const int tid = threadIdx.x;                 // 0..63
  const int p   = blockIdx.x * 64 + tid;       // global (b,t) index
  const int b   = p >> 10;
  const int t   = p & 1023;

  // async gather of this position's h column (strided 4B copies)
  for (int d = 0; d < DD; ++d)
    async_b32(lds_off(&h_lds[tid * 132 + d]), &H[(b * DD + d) * TT + t]);

  // prefetch codebook chunk 0: 128 d rows x 32 k = 1024 x 16B segments
  for (int i = 0; i < 16; ++i) {
    int g = i * 64 + tid;          // 0..1023
    int d = g >> 3, kq = g & 7;
    async_b128(lds_off(&mbuf[0][d * 32 + kq * 4]),
               &M[d * KK + kq * 4]);
  }

  float best  = 3.4e38f;
  int   besti = 0;

  for (int kb = 0; kb < KK / 32; ++kb) {
    wait_async0();
    __syncthreads();               // chunk kb (and h on kb==0) visible block-wide
    if (kb + 1 < KK / 32) {        // prefetch next chunk into the other buffer
      const int kbase = (kb + 1) * 32;
      float* nb = mbuf[(kb + 1) & 1];
      for (int i = 0; i < 16; ++i) {
        int g = i * 64 + tid;
        int d = g >> 3, kq = g & 7;
        async_b128(lds_off(&nb[d * 32 + kq * 4]),
                   &M[d * KK + kbase + kq * 4]);
      }
    }
    const float* mb = mbuf[kb & 1];

    float dist[32];
#pragma unroll
    for (int j = 0; j < 32; ++j) dist[j] = 0.0f;
    for (int dc = 0; dc < DD; dc += 16) {
      float4 hq[4];
#pragma unroll
      for (int q = 0; q < 4; ++q)
        hq[q] = *(const float4*)&h_lds[tid * 132 + dc + q * 4];
#pragma unroll
      for (int q = 0; q < 4; ++q) {
        const float hv[4] = {hq[q].x, hq[q].y, hq[q].z, hq[q].w};
#pragma unroll
        for (int e = 0; e < 4; ++e) {
          const int d = dc + q * 4 + e;
#pragma unroll
          for (int kg = 0; kg < 32; kg += 4) {
            float4 m = *(const float4*)&mb[d * 32 + kg];   // broadcast read
            dist[kg]     += fabsf(hv[e] - m.x);
            dist[kg + 1] += fabsf(hv[e] - m.y);
            dist[kg + 2] += fabsf(hv[e] - m.z);
            dist[kg + 3] += fabsf(hv[e] - m.w);
          }
        }
      }
    }
#pragma unroll
    for (int j = 0; j < 32; ++j) {
      int kg = kb * 32 + j;
      if (dist[j] < best) { best = dist[j]; besti = kg; }
    }
  }

  // squared L2 distance to winning codeword
  float acc = 0.0f;
  for (int d = 0; d < DD; ++d) {
    float diff = h_lds[tid * 132 + d] - M[d * KK + besti];
    acc += diff * diff;
  }
  atomicAdd(&ws[0], acc);
}

// ---------------------------------------------------------------------------
// Decoder GEMM via V_WMMA_F32_16X16X4_F32.
// Block = 64 threads (2 waves), 32 rows of (b,t). Hdec tile async-staged.
// Phase A: Xhat[16x16] tiles = Hdec(16x4) x W^T(4x16) over K=256 -> residual R,
//          partial sums for loss_rec / loss_d.
// Phase B: E[32x256] += R^T(16x4) x Hdec(4x16) over the 32 block rows.
// ---------------------------------------------------------------------------
__global__ void k_dec(const float* __restrict__ X, const float* __restrict__ Hdec,
                      const float* __restrict__ W, const float* __restrict__ wd,
                      float* __restrict__ ws) {
  __shared__ __attribute__((aligned(16))) float hd[32 * 260]; // pad 260 (1040B, 16B-mult)
  __shared__ float rl[32 * 33];    // residual tile (pad 33)
  const int tid  = threadIdx.x;    // 0..63
  const int wave = tid >> 5;       // 0..1
  const int lane = tid & 31;
  const int half = lane >> 4;      // A/B lane half (K 0,1 vs 2,3)
  const int ln   = lane & 15;
  const int row0 = blockIdx.x * 32;

  // Async stage of the Hdec tile [32 x 256]: 2048 x 16B segments
  for (int i = 0; i < 32; ++i) {
    int idx = i * 64 + tid;        // 0..2047
    int r = idx >> 6, g = idx & 63;
    async_b128(lds_off(&hd[r * 260 + g * 4]),
               &Hdec[(row0 + r) * FF + g * 4]);
  }
  wait_async0();
  __syncthreads();

  float rec_acc = 0.0f, d_acc = 0.0f;
  const int rw = wave * 16;        // this wave's 16-row m-tile
  for (int ct = 0; ct < 2; ++ct) {
    const int c0 = ct * 16;
    v8f acc = {0.f,0.f,0.f,0.f,0.f,0.f,0.f,0.f};
    for (int k = 0; k < FF; k += 4) {
      // A: Hdec[row][k + 2*half .. +1]   (16x4, f32 layout)
      v2f a  = *(const v2f*)&hd[(rw + ln) * 260 + k + 2 * half];
      // B: B[kk][n] = W[c0+n][k+kk] -> lane loads W[c0+ln][k+2*half .. +1]
      v2f bv = *(const v2f*)&W[(c0 + ln) * FF + k + 2 * half];
      acc = __builtin_amdgcn_wmma_f32_16x16x4_f32(
          false, a, false, bv, (short)0, acc, false, false);
    }
#pragma unroll
    for (int j = 0; j < 8; ++j) {
      int m    = j + 8 * half;               // C/D layout: VGPR j -> M=j / j+8
      int grow = row0 + rw + m;
      int c    = c0 + ln;
      float xh = acc[j];
      float r  = xh - X[grow * CC + c];
      rec_acc += r * r;
      d_acc   += xh * wd[c];
      rl[(rw + m) * 33 + c] = r;
    }
  }
  atomicAdd(&ws[1], rec_acc);
  atomicAdd(&ws[2], d_acc);
  __syncthreads();

  // Phase B: E(32x256) partial = R^T * Hdec over 32 rows (K-steps of 4)
  for (int pi = 0; pi < 16; ++pi) {
    int tl = pi * 2 + wave;        // 0..31 tile id
    int c0 = (tl & 1) * 16;
    int f0 = (tl >> 1) * 16;
    v8f acc = {0.f,0.f,0.f,0.f,0.f,0.f,0.f,0.f};
    for (int krow = 0; krow < 32; krow += 4) {
      int kr = krow + 2 * half;
      v2f a, bv;
      a.x  = rl[kr * 33 + c0 + ln];          // A[m=c][k=row]
      a.y  = rl[(kr + 1) * 33 + c0 + ln];
      bv.x = hd[kr * 260 + f0 + ln];         // B[k=row][n=f]
      bv.y = hd[(kr + 1) * 260 + f0 + ln];
      acc = __builtin_amdgcn_wmma_f32_16x16x4_f32(
          false, a, false, bv, (short)0, acc, false, false);
    }
#pragma unroll
    for (int j = 0; j < 8; ++j) {
      int c = c0 + j + 8 * half;
      int f = f0 + ln;
      atomicAdd(&ws[WS_E + c * FF + f], acc[j]);
    }
  }
}

// S[f] = sum over all 8192 rows of Hdec[row][f]
__global__ void k_sumf(const float* __restrict__ Hdec, float* __restrict__ ws) {
  const int f  = threadIdx.x;           // 0..255
  const int r0 = blockIdx.x * 256;      // 32 blocks
  float s = 0.0f;
  for (int r = 0; r < 256; ++r) s += Hdec[(r0 + r) * FF + f];
  atomicAdd(&ws[WS_S + f], s);
}

// Combine everything into the scalar loss.
__global__ void k_fin(const float* __restrict__ wd, const float* __restrict__ ws,
                      float* __restrict__ out) {
  __shared__ float red[256];
  const int tid = threadIdx.x;

  float e2 = 0.0f;
  for (int i = tid; i < CC * FF; i += 256) { float v = ws[WS_E + i]; e2 += v * v; }
  red[tid] = e2; __syncthreads();
  for (int s = 128; s > 0; s >>= 1) { if (tid < s) red[tid] += red[tid + s]; __syncthreads(); }
  float sumE2 = red[0]; __syncthreads();

  float s2 = (tid < FF) ? ws[WS_S + tid] * ws[WS_S + tid] : 0.0f;
  red[tid] = s2; __syncthreads();
  for (int s = 128; s > 0; s >>= 1) { if (tid < s) red[tid] += red[tid + s]; __syncthreads(); }
  float sumS2 = red[0]; __syncthreads();

  float w2 = (tid < CC) ? wd[tid] * wd[tid] : 0.0f;
  red[tid] = w2; __syncthreads();
  for (int s = 128; s > 0; s >>= 1) { if (tid < s) red[tid] += red[tid + s]; __syncthreads(); }

  if (tid == 0) {
    float sumW2    = red[0];
    float g_rec_n  = (2.0f / (float)(BT * CC)) * sqrtf(sumE2);
    float g_d_n    = (1.0f / (float)BT) * sqrtf(sumW2) * sqrtf(sumS2);
    float lmbda    = g_rec_n / (g_d_n + 1e-6f);
    float loss_rec = ws[1] / (float)(BT * CC);
    float loss_d   = -ws[2] / (float)BT;
    float loss_m   = 2.0f * ws[0] / (float)(BB * DD * TT);
    out[0] = loss_rec + loss_m + lmbda * loss_d;   // ALPHA = 1
  }
}

extern "C" void kernel_launch(void* const* d_in, const int* in_sizes, int n_in,
                              void* d_out, int out_size, void* d_ws, size_t ws_size,
                              hipStream_t stream) {
  const float* X    = (const float*)d_in[0];
  const float* H    = (const float*)d_in[1];
  const float* M    = (const float*)d_in[2];
  const float* Hdec = (const float*)d_in[3];
  const float* W    = (const float*)d_in[4];
  const float* wd   = (const float*)d_in[5];
  float* ws  = (float*)d_ws;
  float* out = (float*)d_out;

  hipLaunchKernelGGL(k_init, dim3((WS_TOTAL + 255) / 256), dim3(256), 0, stream, ws);
  hipLaunchKernelGGL(k_mem,  dim3(BT / 64),  dim3(64),  0, stream, H, M, ws);
  hipLaunchKernelGGL(k_dec,  dim3(BT / 32),  dim3(64),  0, stream, X, Hdec, W, wd, ws);
  hipLaunchKernelGGL(k_sumf, dim3(BT / 256), dim3(256), 0, stream, Hdec, ws);
  hipLaunchKernelGGL(k_fin,  dim3(1),        dim3(256), 0, stream, wd, ws, out);
}